// NaivePhysicsLoss_51256139710809
// MI455X (gfx1250) — compile-verified
//
#include <hip/hip_runtime.h>
#include <hip/hip_bf16.h>
#include <math.h>

// ---------------- constants from the reference ----------------
#define HID 256
#define KSTEPS (HID / 4)
#define F_Cc 1000.0f
#define M_Cc 500.0f
#define U_Cc 0.01f
#define TH_Cc 0.001f
#define EAc (10000.0f * 0.01f)    // PROP_E * PROP_A
#define EIc (10000.0f * 0.0001f)  // PROP_E * PROP_I22

typedef float v2f __attribute__((ext_vector_type(2)));
typedef float v8f __attribute__((ext_vector_type(8)));

// LDS layout for the fused MLP kernel (dynamic shared memory)
#define S_ROW 260                  // 256 + 4 pad -> conflict-free ds_load_b64 A-fragments
#define S_STREAM (16 * S_ROW)      // 16 node-rows per workgroup
#define S_FLOATS (7 * S_STREAM)    // 7 jet streams
#define OUTP_FLOATS (16 * 16 * 21) // [colTile][row][output] deterministic partials
#define SMEM_FLOATS (S_FLOATS + OUTP_FLOATS + 48)

__device__ __forceinline__ float bfly16(float x) {
  // sum across each 16-lane group of the wave32 (masks < 16 never cross halves)
  x += __shfl_xor(x, 1, 32);
  x += __shfl_xor(x, 2, 32);
  x += __shfl_xor(x, 4, 32);
  x += __shfl_xor(x, 8, 32);
  return x;
}

// One-shot pre-pack of W2 into row-pair-interleaved layout so each WMMA
// B-fragment becomes a single aligned global_load_b64:
//   W2p[(p*HID + j)*2 + {0,1}] = W2[(2p+{0,1})*HID + j]
__global__ void k_packW2(const float* __restrict__ W2, float* __restrict__ W2p) {
  int i = blockIdx.x * blockDim.x + threadIdx.x;  // i in [0, 128*HID)
  if (i >= (HID / 2) * HID) return;
  int p = i / HID, j = i % HID;
  W2p[(size_t)i * 2 + 0] = W2[(size_t)(2 * p + 0) * HID + j];
  W2p[(size_t)i * 2 + 1] = W2[(size_t)(2 * p + 1) * HID + j];
}

// Fused: layer1 + taylor-jet seeds -> 7-stream 256x256 GEMM (f32 WMMA) ->
// tanh jet composition -> W3 contraction -> 21 floats per node:
// [0:15]=pred, 15=dux_dx 16=duz_dz 17=d2uz_dx2 18=d2ux_dz2 19=d3uz_dx3 20=d3ux_dz3
__global__ __launch_bounds__(256) void k_mlp_jet(
    const float* __restrict__ coords, const float* __restrict__ W1,
    const float* __restrict__ b1, const float* __restrict__ W2p,
    const float* __restrict__ b2, const float* __restrict__ W3,
    const float* __restrict__ b3, float* __restrict__ nodeout, int N) {
  extern __shared__ float smem[];
  float* S = smem;                   // [7][16][S_ROW]
  float* outp = smem + S_FLOATS;     // [16][16][21]
  float* clds = outp + OUTP_FLOATS;  // [16][3]

  const int tid = threadIdx.x;
  // ---- phase 0: stage this block's 16 nodes' coords ----
  if (tid < 48) {
    int row = tid / 3, d = tid % 3;
    int node = blockIdx.x * 16 + row;
    clds[tid] = (node < N) ? coords[node * 3 + d] : 0.f;
  }
  __syncthreads();

  // ---- phase 1: z1 = coords@W1+b1 ; h1 = tanh ; jet seeds (u2=u3=0 at input) ----
  // EX tangent into layer1 is W1 row 0, EZ is W1 row 2 (constant across nodes).
#pragma unroll
  for (int i = 0; i < 16; ++i) {
    int e = tid + i * 256;  // 16 rows x 256 k
    int row = e >> 8, k = e & 255;
    float c0 = clds[row * 3 + 0], c1 = clds[row * 3 + 1], c2 = clds[row * 3 + 2];
    float w0 = W1[k], w1 = W1[HID + k], w2 = W1[2 * HID + k];
    float z = fmaf(c0, w0, fmaf(c1, w1, fmaf(c2, w2, b1[k])));
    float T = tanhf(z);
    float T2 = T * T;
    float g1 = 1.f - T2;
    float g2 = -2.f * T * g1;
    float g3 = 2.f * g1 * (3.f * T2 - 1.f);
    float* sp = S + row * S_ROW + k;
    sp[0 * S_STREAM] = T;                  // primal h1
    sp[1 * S_STREAM] = g1 * w0;            // b1x
    sp[2 * S_STREAM] = g2 * w0 * w0;       // b2x
    sp[3 * S_STREAM] = g3 * w0 * w0 * w0;  // b3x
    sp[4 * S_STREAM] = g1 * w2;            // b1z
    sp[5 * S_STREAM] = g2 * w2 * w2;       // b2z
    sp[6 * S_STREAM] = g3 * w2 * w2 * w2;  // b3z
  }
  __syncthreads();

  // ---- phase 2: 7-stream GEMM vs W2 with V_WMMA_F32_16X16X4_F32 ----
  const int lane = tid & 31;
  const int wave = tid >> 5;
  const int rowA = lane & 15;         // A row / B,C,D column within tile
  const int koff = (lane >> 4) << 1;  // A 16x4 layout: v0=K0|K2, v1=K1|K3 per half-wave
  const int phalf = lane >> 4;        // which W2 row-pair (k 0-1 or 2-3 of the step) this half-wave needs
  v8f acc[7][2];
#pragma unroll
  for (int s = 0; s < 7; ++s)
#pragma unroll
    for (int t = 0; t < 2; ++t) acc[s][t] = (v8f){0.f, 0.f, 0.f, 0.f, 0.f, 0.f, 0.f, 0.f};

  const int j0 = (wave * 2 + 0) * 16 + rowA;  // wave owns 2 output column tiles
  const int j1 = (wave * 2 + 1) * 16 + rowA;

  for (int kk = 0; kk < KSTEPS; ++kk) {
    const int kb = (kk << 2) + koff;
    const int ppair = (kk << 1) + phalf;  // row-pair index into packed W2
    v2f af[7];
#pragma unroll
    for (int s = 0; s < 7; ++s)
      af[s] = *(const v2f*)(S + s * S_STREAM + rowA * S_ROW + kb);  // ds_load_b64
    // single global_load_b64 per B-fragment (packed row pair, L2/WGP$-hot)
    v2f bf0 = *(const v2f*)(W2p + ((size_t)ppair * HID + j0) * 2);
    v2f bf1 = *(const v2f*)(W2p + ((size_t)ppair * HID + j1) * 2);
#pragma unroll
    for (int s = 0; s < 7; ++s) {
      acc[s][0] = __builtin_amdgcn_wmma_f32_16x16x4_f32(false, af[s], false, bf0,
                                                        (short)0, acc[s][0], false, false);
      acc[s][1] = __builtin_amdgcn_wmma_f32_16x16x4_f32(false, af[s], false, bf1,
                                                        (short)0, acc[s][1], false, false);
    }
  }

  // ---- phase 3: tanh jet composition + W3 contraction (butterfly over j-lanes) ----
#pragma unroll
  for (int t = 0; t < 2; ++t) {
    const int jt = wave * 2 + t;
    const int j = jt * 16 + rowA;
    const float b2j = b2[j];
    float w3c[15];
#pragma unroll
    for (int o = 0; o < 15; ++o) w3c[o] = W3[j * 15 + o];
#pragma unroll
    for (int v = 0; v < 8; ++v) {  // C/D layout: vgpr v -> row v (lanes<16) / v+8 (lanes>=16)
      const int row = v + ((lane >> 4) << 3);
      float z = acc[0][t][v] + b2j;
      float T = tanhf(z);
      float T2 = T * T;
      float g1 = 1.f - T2;
      float g2 = -2.f * T * g1;
      float g3 = 2.f * g1 * (3.f * T2 - 1.f);
      float y1x = acc[1][t][v], y2x = acc[2][t][v], y3x = acc[3][t][v];
      float y1z = acc[4][t][v], y2z = acc[5][t][v], y3z = acc[6][t][v];
      float A1x = g1 * y1x;
      float A2x = g2 * y1x * y1x + g1 * y2x;
      float A3x = g3 * y1x * y1x * y1x + 3.f * g2 * y1x * y2x + g1 * y3x;
      float A1z = g1 * y1z;
      float A2z = g2 * y1z * y1z + g1 * y2z;
      float A3z = g3 * y1z * y1z * y1z + 3.f * g2 * y1z * y2z + g1 * y3z;
      float* op = outp + (jt * 16 + row) * 21;
      float c;
#pragma unroll
      for (int o = 0; o < 15; ++o) {
        c = bfly16(T * w3c[o]);
        if ((lane & 15) == 0) op[o] = c;
      }
      c = bfly16(A1x * w3c[0]); if ((lane & 15) == 0) op[15] = c;  // dux_dx
      c = bfly16(A1z * w3c[1]); if ((lane & 15) == 0) op[16] = c;  // duz_dz
      c = bfly16(A2x * w3c[1]); if ((lane & 15) == 0) op[17] = c;  // d2uz_dx2
      c = bfly16(A2z * w3c[0]); if ((lane & 15) == 0) op[18] = c;  // d2ux_dz2
      c = bfly16(A3x * w3c[1]); if ((lane & 15) == 0) op[19] = c;  // d3uz_dx3
      c = bfly16(A3z * w3c[0]); if ((lane & 15) == 0) op[20] = c;  // d3ux_dz3
    }
  }
  __syncthreads();

  // ---- phase 4: deterministic fixed-order sum over the 16 column tiles ----
  const int base = blockIdx.x * 16;
  for (int tt = tid; tt < 16 * 21; tt += blockDim.x) {
    int row = tt / 21, o = tt % 21;
    float s = (o < 15) ? b3[o] : 0.f;
#pragma unroll
    for (int jt = 0; jt < 16; ++jt) s += outp[(jt * 16 + row) * 21 + o];
    int node = base + row;
    if (node < N) nodeout[(size_t)node * 21 + o] = s;
  }
}

__global__ void k_zero(float* __restrict__ p, int n) {
  int i = blockIdx.x * blockDim.x + threadIdx.x;
  if (i < n) p[i] = 0.f;
}

// ff_A / ff_B scatter per the reference's .at[].set(mode='drop')
__global__ void k_scatter(const float* __restrict__ nodeout,
                          const float* __restrict__ face_mask,
                          const int* __restrict__ feid, const int* __restrict__ fae,
                          float* __restrict__ ffA, float* __restrict__ ffB, int N, int E) {
  int i = blockIdx.x * blockDim.x + threadIdx.x;
  if (i >= N * 4) return;
  int n = i >> 2, f = i & 3;
  if (face_mask[i] > 0.5f) {
    int e = feid[i];
    if (e >= 0 && e < E) {
      const float* fp = nodeout + (size_t)n * 21 + 3 + f * 3;
      float* dst = (fae[i] == 1) ? ffA : ffB;
      dst[e * 3 + 0] = fp[0];
      dst[e * 3 + 1] = fp[1];
      dst[e * 3 + 2] = fp[2];
    }
  }
}

// per-block deterministic partial sums of the 12 reduction quantities
__global__ __launch_bounds__(256) void k_reduce(
    const float* __restrict__ nodeout, const float* __restrict__ F_ext,
    const float* __restrict__ bc_disp, const float* __restrict__ bc_rot,
    const float* __restrict__ face_mask, const float* __restrict__ ffA,
    const float* __restrict__ ffB, const float* __restrict__ dirs,
    const int* __restrict__ conn, float* __restrict__ partials, int N, int E) {
  float a[12];
#pragma unroll
  for (int q = 0; q < 12; ++q) a[q] = 0.f;
  const int stride = gridDim.x * blockDim.x;
  const int nmax = (N > E) ? N : E;
  for (int i = blockIdx.x * blockDim.x + threadIdx.x; i < nmax; i += stride) {
    if (i < N) {
      const float* no = nodeout + (size_t)i * 21;
      float r0 = no[3] + no[6] + no[9] + no[12] - F_ext[i * 3 + 0];
      float r1 = no[4] + no[7] + no[10] + no[13] - F_ext[i * 3 + 1];
      float r2 = no[5] + no[8] + no[11] + no[14] - F_ext[i * 3 + 2];
      if (bc_disp[i] < 0.5f) {  // L_eq (free nodes)
        a[0] += (r0 / F_Cc) * (r0 / F_Cc) + (r1 / F_Cc) * (r1 / F_Cc) +
                (r2 / M_Cc) * (r2 / M_Cc);
        a[1] += 1.f;
      }
#pragma unroll
      for (int f = 0; f < 4; ++f) {  // L_free (mask < 0.5 faces)
        if (face_mask[i * 4 + f] < 0.5f) {
          float f0 = no[3 + f * 3] / F_Cc, f1 = no[4 + f * 3] / F_Cc,
                f2 = no[5 + f * 3] / M_Cc;
          a[2] += f0 * f0 + f1 * f1 + f2 * f2;
          a[3] += 3.f;  // mask broadcast over 3 components
        }
      }
      if (bc_disp[i] > 0.5f) {  // L_sup (disp comps)
        float d0 = no[0] / U_Cc, d1 = no[1] / U_Cc;
        a[4] += d0 * d0;
        a[5] += d1 * d1;
        a[6] += 1.f;
      }
      if (bc_rot[i] > 0.5f) {  // L_sup (rot comp)
        float d2 = no[2] / TH_Cc;
        a[7] += d2 * d2;
        a[8] += 1.f;
      }
    }
    if (i < E) {
      float ca = dirs[i * 3 + 0], sa = dirs[i * 3 + 2];
      bool horiz = fabsf(ca) > fabsf(sa);
      float fa0 = ffA[i * 3], fa1 = ffA[i * 3 + 1], fa2 = ffA[i * 3 + 2];
      float fb0 = ffB[i * 3], fb1 = ffB[i * 3 + 1], fb2 = ffB[i * 3 + 2];
      float fal0 = fa0 * ca + fa1 * sa, fal1 = -fa0 * sa + fa1 * ca;
      float fbl0 = fb0 * ca + fb1 * sa, fbl1 = -fb0 * sa + fb1 * ca;
      int nA = conn[i * 2 + 0], nB = conn[i * 2 + 1];
      const float* dA = nodeout + (size_t)nA * 21 + 15;
      const float* dB = nodeout + (size_t)nB * 21 + 15;
      float strA = horiz ? dA[0] : dA[1];
      float strB = horiz ? dB[0] : dB[1];
      float rA = (fal0 + EAc * strA) / F_Cc, rB = (fbl0 - EAc * strB) / F_Cc;
      a[9] += rA * rA + rB * rB;  // L_N
      float cvA = horiz ? dA[2] : -dA[3];
      float cvB = horiz ? dB[2] : -dB[3];
      rA = (fa2 + EIc * cvA) / M_Cc;
      rB = (fb2 - EIc * cvB) / M_Cc;
      a[10] += rA * rA + rB * rB;  // L_M
      float wA = horiz ? dA[4] : -dA[5];
      float wB = horiz ? dB[4] : -dB[5];
      rA = (fal1 + EIc * wA) / F_Cc;
      rB = (fbl1 - EIc * wB) / F_Cc;
      a[11] += rA * rA + rB * rB;  // L_V
    }
  }
  __shared__ float sb[256];
  for (int q = 0; q < 12; ++q) {
    sb[threadIdx.x] = a[q];
    __syncthreads();
    for (int s = 128; s > 0; s >>= 1) {
      if (threadIdx.x < s) sb[threadIdx.x] += sb[threadIdx.x + s];
      __syncthreads();
    }
    if (threadIdx.x == 0) partials[blockIdx.x * 12 + q] = sb[0];
    __syncthreads();
  }
}

__global__ void k_final(const float* __restrict__ partials, int nb, float Ef,
                        float* __restrict__ out) {
  __shared__ float tot[12];
  int q = threadIdx.x;
  if (q < 12) {
    float s = 0.f;
    for (int b = 0; b < nb; ++b) s += partials[b * 12 + q];  // fixed order
    tot[q] = s;
  }
  __syncthreads();
  if (q == 0) {
    float L_eq = tot[0] / fmaxf(tot[1], 1.f);
    float L_free = tot[2] / fmaxf(tot[3], 1.f);
    float L_sup = tot[4] / fmaxf(tot[6], 1.f) + tot[5] / fmaxf(tot[6], 1.f) +
                  tot[7] / fmaxf(tot[8], 1.f);
    out[0] = L_eq + L_free + L_sup + tot[9] / Ef + tot[10] / Ef + tot[11] / Ef;
  }
}

extern "C" void kernel_launch(void* const* d_in, const int* in_sizes, int n_in,
                              void* d_out, int out_size, void* d_ws, size_t ws_size,
                              hipStream_t stream) {
  const float* coords = (const float*)d_in[0];
  const float* W1 = (const float*)d_in[1];
  const float* b1 = (const float*)d_in[2];
  const float* W2 = (const float*)d_in[3];
  const float* b2 = (const float*)d_in[4];
  const float* W3 = (const float*)d_in[5];
  const float* b3 = (const float*)d_in[6];
  const float* dirs = (const float*)d_in[7];
  const float* F_ext = (const float*)d_in[8];
  const float* bc_disp = (const float*)d_in[9];
  const float* bc_rot = (const float*)d_in[10];
  const float* face_mask = (const float*)d_in[11];
  const int* conn = (const int*)d_in[12];
  const int* feid = (const int*)d_in[13];
  const int* fae = (const int*)d_in[14];
  const int N = in_sizes[0] / 3;
  const int E = in_sizes[12] / 2;

  float* nodeout = (float*)d_ws;                  // N*21 floats
  float* ffA = nodeout + (size_t)N * 21;          // E*3
  float* ffB = ffA + (size_t)E * 3;               // E*3 (contiguous with ffA)
  float* partials = ffB + (size_t)E * 3;          // NB*12
  const int NB = 256;
  float* W2p = partials + (size_t)NB * 12;        // HID*HID packed row pairs

  k_packW2<<<((HID / 2) * HID + 255) / 256, 256, 0, stream>>>(W2, W2p);

  const int nblk = (N + 15) / 16;
  const size_t smem = (size_t)SMEM_FLOATS * sizeof(float);  // ~138 KB (<320 KB/WGP)
  k_mlp_jet<<<nblk, 256, smem, stream>>>(coords, W1, b1, W2p, b2, W3, b3, nodeout, N);
  const int zn = E * 6;
  k_zero<<<(zn + 255) / 256, 256, 0, stream>>>(ffA, zn);
  k_scatter<<<(N * 4 + 255) / 256, 256, 0, stream>>>(nodeout, face_mask, feid, fae,
                                                     ffA, ffB, N, E);
  k_reduce<<<NB, 256, 0, stream>>>(nodeout, F_ext, bc_disp, bc_rot, face_mask, ffA,
                                   ffB, dirs, conn, partials, N, E);
  k_final<<<1, 64, 0, stream>>>(partials, NB, (float)E, (float*)d_out);
}